// MambaLayer_67319317397651
// MI455X (gfx1250) — compile-verified
//
#include <hip/hip_runtime.h>
#include <hip/hip_bf16.h>

// ---- problem constants (match reference) ----
#define BB 2
#define DM 192
#define DI 384
#define NN 16
#define RR 12
#define KK 8
#define HH 48
#define WW 48
#define LL (HH*WW)      // 2304
#define E2 (2*DI)       // 768
#define C2N (RR + 2*NN) // 44
#define CDL 48          // padded x_dbl row count (multiple of 16)

typedef float v2f __attribute__((ext_vector_type(2)));
typedef float v8f __attribute__((ext_vector_type(8)));

// CDNA5 fp32 WMMA: D(16x16) = A(16x4) * B(4x16) + C, full fp32 precision.
__device__ __forceinline__ v8f wmma_f32(v2f a, v2f b, v8f c) {
    return __builtin_amdgcn_wmma_f32_16x16x4_f32(
        /*neg_a=*/false, a, /*neg_b=*/false, b,
        /*c_mod=*/(short)0, c, /*reuse_a=*/false, /*reuse_b=*/false);
}

// scan_src: scan position p of direction k reads spatial index s of xc
__device__ __forceinline__ int scan_src(int k, int p) {
    int pp  = (k & 2) ? (LL - 1 - p) : p;
    int fam = ((k & 4) >> 1) | (k & 1); // 0=id 1=transpose 2=diag 3=anti
    if (fam == 0) return pp;
    int wq = pp / HH, h = pp % HH;
    if (fam == 1) return h * WW + wq;
    if (fam == 2) return h * WW + ((h + wq) % WW);
    int t = (wq - h) % WW; if (t < 0) t += WW;
    return h * WW + t;
}

// merge_src: spatial index s receives scan output of direction k at position p
__device__ __forceinline__ int merge_src(int k, int s) {
    int h = s / WW, w = s % WW;
    int fam = ((k & 4) >> 1) | (k & 1);
    int p;
    if (fam == 0)      p = s;
    else if (fam == 1) p = w * HH + h;
    else if (fam == 2) { int t = (w - h) % WW; if (t < 0) t += WW; p = t * HH + h; }
    else               { int t = (w + h) % WW; p = t * HH + h; }
    if (k & 2) p = LL - 1 - p;
    return p;
}

// ---- Stage 1: in_proj GEMM: xz[b,e,l] = sum_c W[e,c]*x[b,c,l] + bias[e] ----
__global__ __launch_bounds__(32) void k_in_proj(const float* __restrict__ x,
                                                const float* __restrict__ w,
                                                const float* __restrict__ bias,
                                                float* __restrict__ xz) {
    int lane = threadIdx.x;
    int n0 = blockIdx.x * 16, m0 = blockIdx.y * 16, b = blockIdx.z;
    int mrow = m0 + (lane & 15);
    int kg   = (lane >> 4) * 2;
    int ncol = lane & 15;
    const float* X = x + (size_t)b * DM * LL;
    v8f acc = {};
    for (int k0 = 0; k0 < DM; k0 += 4) {
        v2f a, bb;
        a.x  = w[mrow * DM + k0 + kg];
        a.y  = w[mrow * DM + k0 + kg + 1];
        bb.x = X[(size_t)(k0 + kg) * LL + n0 + ncol];
        bb.y = X[(size_t)(k0 + kg + 1) * LL + n0 + ncol];
        acc = wmma_f32(a, bb, acc);
    }
    float* out = xz + (size_t)b * E2 * LL;
    int rbase = m0 + (lane >> 4) * 8;
#pragma unroll
    for (int i = 0; i < 8; i++) {
        int row = rbase + i;
        out[(size_t)row * LL + n0 + ncol] = acc[i] + bias[row];
    }
}

// ---- Stage 2: depthwise 3x3 conv (SAME) + bias + SiLU -> xcT (B, L, DI) ----
__global__ void k_conv_silu(const float* __restrict__ xz,
                            const float* __restrict__ cw,
                            const float* __restrict__ cb,
                            float* __restrict__ xcT) {
    int idx = blockIdx.x * blockDim.x + threadIdx.x;
    if (idx >= BB * DI * LL) return;
    int b = idx / (DI * LL);
    int r = idx % (DI * LL);
    int d = r / LL;
    int l = r % LL;
    int h = l / WW, w = l % WW;
    const float* xi = xz + (size_t)b * E2 * LL + (size_t)d * LL;
    const float* wt = cw + d * 9;
    float s = cb[d];
#pragma unroll
    for (int i = 0; i < 3; i++) {
        int hy = h + i - 1;
        if (hy < 0 || hy >= HH) continue;
#pragma unroll
        for (int j = 0; j < 3; j++) {
            int wx = w + j - 1;
            if (wx < 0 || wx >= WW) continue;
            s += wt[i * 3 + j] * xi[hy * WW + wx];
        }
    }
    float sil = s / (1.f + expf(-s));
    xcT[(size_t)b * LL * DI + (size_t)l * DI + d] = sil;
}

// ---- Stage 3: x_proj GEMM with gathered columns (cross-scan folded in) ----
// xdbl[(b,k), l, c] = sum_d x_proj_w[k,c,d] * xcT[b, perm_k(l), d]
__global__ __launch_bounds__(32) void k_xproj(const float* __restrict__ xcT,
                                              const float* __restrict__ xpw,
                                              float* __restrict__ xdbl) {
    int lane = threadIdx.x;
    int n0 = blockIdx.x * 16, m0 = blockIdx.y * 16;
    int bk = blockIdx.z;
    int b = bk >> 3, k = bk & 7;
    int mrow = m0 + (lane & 15);
    int kg   = (lane >> 4) * 2;
    int ncol = lane & 15;
    const float* A = xpw + (size_t)k * C2N * DI;
    int ar = mrow < C2N ? mrow : (C2N - 1);   // clamp: garbage rows never stored
    int scol = scan_src(k, n0 + ncol);
    const float* Bc = xcT + (size_t)b * LL * DI + (size_t)scol * DI;
    v8f acc = {};
    for (int k0 = 0; k0 < DI; k0 += 4) {
        v2f a, bb;
        a.x  = A[(size_t)ar * DI + k0 + kg];
        a.y  = A[(size_t)ar * DI + k0 + kg + 1];
        bb.x = Bc[k0 + kg];
        bb.y = Bc[k0 + kg + 1];
        acc = wmma_f32(a, bb, acc);
    }
    float* out = xdbl + ((size_t)bk * LL + n0 + ncol) * CDL;
    int rbase = m0 + (lane >> 4) * 8;
#pragma unroll
    for (int i = 0; i < 8; i++) {
        int row = rbase + i;
        if (row < C2N) out[row] = acc[i];
    }
}

// ---- Stage 4: fused dt-proj + softplus + selective scan ----
// one thread per (b,k,d); 2304 sequential steps.
// The 48-float x_dbl column for step p+PF is prefetched into a 5-slot LDS ring
// with GLOBAL_LOAD_ASYNC_TO_LDS_B128 (ASYNCcnt-tracked), so each step's column
// is already LDS-resident: step latency = barrier + LDS read + ALU.
#define SCAN_T 128
#define SCAN_G (DI / SCAN_T)  // 3
#define PF 4                  // async pipeline depth (columns in flight)
#define RING (PF + 1)         // ring slots; never overwrites the slot being read
__global__ __launch_bounds__(SCAN_T) void k_scan(const float* __restrict__ xdbl,
                                                 const float* __restrict__ xcT,
                                                 const float* __restrict__ dtw_g,
                                                 const float* __restrict__ dtb_g,
                                                 const float* __restrict__ alog,
                                                 const float* __restrict__ Dg,
                                                 float* __restrict__ yT) {
    int bid = blockIdx.x;
    int g  = bid % SCAN_G;
    int bk = bid / SCAN_G;
    int b = bk >> 3, k = bk & 7;
    int d = g * SCAN_T + threadIdx.x;

    float dtw[RR];
#pragma unroll
    for (int r = 0; r < RR; r++) dtw[r] = dtw_g[((size_t)k * DI + d) * RR + r];
    float dtb = dtb_g[k * DI + d];
    float Arow[NN];
#pragma unroll
    for (int n = 0; n < NN; n++) Arow[n] = -expf(alog[((size_t)k * DI + d) * NN + n]);
    float Dv = Dg[k * DI + d];
    float hst[NN];
#pragma unroll
    for (int n = 0; n < NN; n++) hst[n] = 0.f;

    __shared__ __align__(16) float cbuf[RING][CDL];
    const float* xd = xdbl + (size_t)bk * LL * CDL;
    const float* xc = xcT + (size_t)b * LL * DI;
    float* yo = yT + (size_t)bk * LL * DI;

    // raw LDS byte offset: flat shared-aperture addresses use addr[31:0] as the
    // DS address (ISA 10.2), so the low 32 bits of the generic pointer suffice.
    unsigned lds0 = (unsigned)(unsigned long long)(const void*)&cbuf[0][0];
    // 12 lanes x B128 = one full padded 192B column per issued step
    bool issuer = (threadIdx.x < (CDL * 4) / 16);
    unsigned lane_b = threadIdx.x * 16u;

    // prologue: launch columns 0..PF-1
#pragma unroll
    for (int p = 0; p < PF; p++) {
        if (issuer) {
            unsigned lofs = lds0 + (unsigned)((p % RING) * CDL * 4) + lane_b;
            unsigned long long ga = (unsigned long long)(const char*)xd
                                  + (unsigned long long)p * (CDL * 4) + lane_b;
            asm volatile("global_load_async_to_lds_b128 %0, %1, off"
                         :: "v"(lofs), "v"(ga) : "memory");
        }
    }

    for (int p = 0; p < LL; p++) {
        // ensure column p has landed (async completes in order)
        if (p + PF >= LL) {
            asm volatile("s_wait_asynccnt 0x0" ::: "memory");
        } else {
            asm volatile("s_wait_asynccnt 0x3" ::: "memory");
        }
        __syncthreads();
        const float* cb = cbuf[p % RING];

        float dt = dtb;
#pragma unroll
        for (int r = 0; r < RR; r++) dt += dtw[r] * cb[r];
        dt = (dt > 20.f) ? dt : log1pf(expf(dt));   // softplus
        int s = scan_src(k, p);
        float u  = xc[(size_t)s * DI + d];
        // prefetch the u-gather line a few steps ahead (global_prefetch_b8)
        {
            int pf = p + 8; if (pf >= LL) pf = LL - 1;
            __builtin_prefetch(xc + (size_t)scan_src(k, pf) * DI + d, 0, 0);
        }
        float du = dt * u;
        float y = 0.f;
#pragma unroll
        for (int n = 0; n < NN; n++) {
            float dA = expf(dt * Arow[n]);
            hst[n] = hst[n] * dA + du * cb[RR + n];        // B column
            y += hst[n] * cb[RR + NN + n];                 // C column
        }
        // launch column p+PF into ring slot (p+PF)%RING (never the slot read now)
        if (issuer && (p + PF) < LL) {
            int pn = p + PF;
            unsigned lofs = lds0 + (unsigned)((pn % RING) * CDL * 4) + lane_b;
            unsigned long long ga = (unsigned long long)(const char*)xd
                                  + (unsigned long long)pn * (CDL * 4) + lane_b;
            asm volatile("global_load_async_to_lds_b128 %0, %1, off"
                         :: "v"(lofs), "v"(ga) : "memory");
        }
        yo[(size_t)p * DI + d] = y + u * Dv;
    }
}

// ---- Stage 5: cross-merge (inverse perms) + LayerNorm + SiLU(z) gate ----
__global__ __launch_bounds__(DI) void k_merge_ln_gate(const float* __restrict__ yT,
                                                      const float* __restrict__ xz,
                                                      const float* __restrict__ lng,
                                                      const float* __restrict__ lnb,
                                                      float* __restrict__ yg) {
    int b = blockIdx.x / LL;
    int s = blockIdx.x % LL;
    int d = threadIdx.x;
    float m = 0.f;
#pragma unroll
    for (int k = 0; k < KK; k++) {
        int p = merge_src(k, s);
        m += yT[(((size_t)(b * KK + k)) * LL + p) * DI + d];
    }
    // block reduction for mean / variance over DI=384 channels
    float v = m, v2 = m * m;
#pragma unroll
    for (int off = 16; off > 0; off >>= 1) {
        v  += __shfl_down(v,  off, 32);
        v2 += __shfl_down(v2, off, 32);
    }
    __shared__ float red[2][DI / 32];
    __shared__ float stats[2];
    int wid = threadIdx.x >> 5;
    if ((threadIdx.x & 31) == 0) { red[0][wid] = v; red[1][wid] = v2; }
    __syncthreads();
    if (threadIdx.x == 0) {
        float sA = 0.f, sB = 0.f;
        for (int i = 0; i < DI / 32; i++) { sA += red[0][i]; sB += red[1][i]; }
        float mu = sA / (float)DI;
        stats[0] = mu;
        stats[1] = sB / (float)DI - mu * mu;
    }
    __syncthreads();
    float mu = stats[0], var = stats[1];
    float yt = (m - mu) * rsqrtf(var + 1e-5f) * lng[d] + lnb[d];
    float zv = xz[(size_t)b * E2 * LL + (size_t)(DI + d) * LL + s];
    float sil = zv / (1.f + expf(-zv));
    yg[((size_t)b * LL + s) * DI + d] = yt * sil;
}

// ---- Stage 6: out_proj GEMM: out[b,c,l] = sum_d W[c,d]*yg[b,l,d] + bias[c] ----
__global__ __launch_bounds__(32) void k_out_proj(const float* __restrict__ yg,
                                                 const float* __restrict__ w,
                                                 const float* __restrict__ bias,
                                                 float* __restrict__ out) {
    int lane = threadIdx.x;
    int n0 = blockIdx.x * 16, m0 = blockIdx.y * 16, b = blockIdx.z;
    int mrow = m0 + (lane & 15);
    int kg   = (lane >> 4) * 2;
    int ncol = lane & 15;
    const float* Bc = yg + ((size_t)b * LL + n0 + ncol) * DI;
    v8f acc = {};
    for (int k0 = 0; k0 < DI; k0 += 4) {
        v2f a, bb;
        a.x  = w[(size_t)mrow * DI + k0 + kg];
        a.y  = w[(size_t)mrow * DI + k0 + kg + 1];
        bb.x = Bc[k0 + kg];
        bb.y = Bc[k0 + kg + 1];
        acc = wmma_f32(a, bb, acc);
    }
    float* O = out + (size_t)b * DM * LL;
    int rbase = m0 + (lane >> 4) * 8;
#pragma unroll
    for (int i = 0; i < 8; i++) {
        int row = rbase + i;
        O[(size_t)row * LL + n0 + ncol] = acc[i] + bias[row];
    }
}

extern "C" void kernel_launch(void* const* d_in, const int* in_sizes, int n_in,
                              void* d_out, int out_size, void* d_ws, size_t ws_size,
                              hipStream_t stream) {
    (void)in_sizes; (void)n_in; (void)out_size; (void)ws_size;
    const float* x          = (const float*)d_in[0];
    const float* in_proj_w  = (const float*)d_in[1];
    const float* in_proj_b  = (const float*)d_in[2];
    const float* conv_w     = (const float*)d_in[3];
    const float* conv_b     = (const float*)d_in[4];
    const float* x_proj_w   = (const float*)d_in[5];
    const float* dt_w       = (const float*)d_in[6];
    const float* dt_b       = (const float*)d_in[7];
    const float* A_log      = (const float*)d_in[8];
    const float* Ds         = (const float*)d_in[9];
    const float* ln_g       = (const float*)d_in[10];
    const float* ln_b       = (const float*)d_in[11];
    const float* out_proj_w = (const float*)d_in[12];
    const float* out_proj_b = (const float*)d_in[13];
    float* out = (float*)d_out;

    // workspace carve-out (total ~92 MB, all fp32, fully overwritten each call)
    float* ws   = (float*)d_ws;
    float* xz   = ws;                                   // B*768*L
    float* xcT  = xz   + (size_t)BB * E2 * LL;          // B*L*DI
    float* xdbl = xcT  + (size_t)BB * LL * DI;          // B*K*L*48
    float* yT   = xdbl + (size_t)BB * KK * LL * CDL;    // B*K*L*DI
    float* yg   = yT   + (size_t)BB * KK * LL * DI;     // B*L*DI

    // 1) in_proj
    k_in_proj<<<dim3(LL / 16, E2 / 16, BB), 32, 0, stream>>>(x, in_proj_w, in_proj_b, xz);
    // 2) depthwise conv + SiLU -> transposed activation
    {
        int total = BB * DI * LL;
        k_conv_silu<<<(total + 255) / 256, 256, 0, stream>>>(xz, conv_w, conv_b, xcT);
    }
    // 3) x_proj with cross-scan gather folded in
    k_xproj<<<dim3(LL / 16, CDL / 16, BB * KK), 32, 0, stream>>>(xcT, x_proj_w, xdbl);
    // 4) fused dt-projection + softplus + selective scan (async LDS pipeline)
    k_scan<<<BB * KK * SCAN_G, SCAN_T, 0, stream>>>(xdbl, xcT, dt_w, dt_b, A_log, Ds, yT);
    // 5) cross-merge + LayerNorm + gate
    k_merge_ln_gate<<<BB * LL, DI, 0, stream>>>(yT, xz, ln_g, ln_b, yg);
    // 6) out_proj
    k_out_proj<<<dim3(LL / 16, DM / 16, BB), 32, 0, stream>>>(yg, out_proj_w, out_proj_b, out);
}